// SSIM_64587718197595
// MI455X (gfx1250) — compile-verified
//
#include <hip/hip_runtime.h>

// ---------------------------------------------------------------------------
// Fused SSIM for (16,3,512,512) fp32 pairs on gfx1250 (MI455X).
// Separable 11-tap Gaussian blur expressed as Toeplitz GEMMs on the matrix
// pipe via V_WMMA_F32_16X16X4_F32 (wave32). One wave per 16x16 output tile.
// ---------------------------------------------------------------------------

typedef __attribute__((ext_vector_type(2))) float v2f;
typedef __attribute__((ext_vector_type(8))) float v8f;

#define IMG_H   512
#define IMG_W   512
#define NPLANES 48           // 16 * 3
#define RAD     5            // (11-1)/2
#define PCOLS   28           // 26 halo cols padded to K-multiple of 4
#define C1_SSIM 1.0e-4f      // 0.01^2
#define C2_SSIM 9.0e-4f      // 0.03^2

// Normalized 1-D Gaussian, sigma = 1.5, window 11 (matches numpy reference).
__constant__ float c_wg[11] = {
    0.00102838f, 0.00759894f, 0.03600077f, 0.10936073f, 0.21300553f,
    0.26601172f,
    0.21300553f, 0.10936073f, 0.03600077f, 0.00759894f, 0.00102838f
};

__device__ __forceinline__ float prodv(int p, float x, float y) {
    switch (p) {
        case 0:  return x;
        case 1:  return y;
        case 2:  return x * x;
        case 3:  return y * y;
        default: return x * y;
    }
}

__global__ void __launch_bounds__(32)
ssim_zero_kernel(float* out) { out[0] = 0.0f; }

__global__ void __launch_bounds__(32)
ssim_wmma_kernel(const float* __restrict__ img1,
                 const float* __restrict__ img2,
                 float* __restrict__ out)
{
    // LDS: input patches (rows 0..25 real, 26..31 zero; cols 0..25 real) and
    // the horizontal-blur intermediate H (stride 20 avoids bank conflicts
    // between the two 16-lane halves on D-store / B-load).
    __shared__ float sX[32][PCOLS];
    __shared__ float sY[32][PCOLS];
    __shared__ float sH[32][20];

    const int lane  = threadIdx.x;
    const int bid   = blockIdx.x;
    const int plane = bid >> 10;          // 48 planes
    const int t     = bid & 1023;         // 32x32 tiles per plane
    const int trow  = t >> 5;
    const int tcol  = t & 31;
    const int row0  = trow * 16 - RAD;    // patch origin in the image
    const int col0  = tcol * 16 - RAD;

    const float* p1 = img1 + (size_t)plane * (IMG_H * IMG_W);
    const float* p2 = img2 + (size_t)plane * (IMG_H * IMG_W);

    // ---- Stage 26x26 patch (zero padded) into LDS, lane = column ----------
    {
        const int  gc  = col0 + lane;
        const bool cok = (lane < 26) && (gc >= 0) && (gc < IMG_W);
        for (int r = 0; r < 32; ++r) {
            float x = 0.0f, y = 0.0f;
            const int gr = row0 + r;
            if (r < 26 && cok && gr >= 0 && gr < IMG_H) {
                const size_t off = (size_t)gr * IMG_W + gc;
                x = p1[off];
                y = p2[off];
            }
            if (lane < PCOLS) { sX[r][lane] = x; sY[r][lane] = y; }
        }
    }
    __syncthreads();

    // ---- Per-lane Toeplitz weight chunks ----------------------------------
    // A(16x4) layout: lanes 0-15 hold K=0,1 ; lanes 16-31 hold K=2,3 (M=lane&15)
    // B(4x16) layout: VGPR v holds K=(lane<16?0:2)+v, N=lane&15
    // Horizontal B[k][n] = w[k-n] and vertical A[m][k] = w[k-m] coincide per
    // lane, so one set of 7 v2f constants serves both passes.
    const int half = (lane < 16) ? 0 : 2;
    const int nm   = lane & 15;
    v2f wchunk[7];
#pragma unroll
    for (int kc = 0; kc < 7; ++kc) {
#pragma unroll
        for (int v = 0; v < 2; ++v) {
            const int d = 4 * kc + half + v - nm;
            float w = 0.0f;
            if (d >= 0 && d <= 10) w = c_wg[d];
            wchunk[kc][v] = w;
        }
    }

    // ---- Five blurred quantities: mu1, mu2, E[x^2], E[y^2], E[xy] ---------
    v8f acc[5];
#pragma unroll
    for (int p = 0; p < 5; ++p) {
        // Horizontal pass: H(26x16) = Patch(26x28) * Wt(28x16), two 16-row
        // chunks of 7 K=4 WMMA steps each. Rows 26..31 are zero by
        // construction; their vertical weights are zero too.
#pragma unroll
        for (int rc = 0; rc < 2; ++rc) {
            v8f h = {};
            const int row = rc * 16 + nm;
#pragma unroll
            for (int kc = 0; kc < 7; ++kc) {
                const int k0 = 4 * kc + half;
                const float x0 = sX[row][k0],     y0 = sY[row][k0];
                const float x1 = sX[row][k0 + 1], y1 = sY[row][k0 + 1];
                v2f a;
                a[0] = prodv(p, x0, y0);
                a[1] = prodv(p, x1, y1);
                h = __builtin_amdgcn_wmma_f32_16x16x4_f32(
                        false, a, false, wchunk[kc], (short)0, h, false, false);
            }
            // D layout: VGPR j -> row j (lanes 0-15) / row j+8 (lanes 16-31)
            const int srow = rc * 16 + ((lane < 16) ? 0 : 8);
#pragma unroll
            for (int j = 0; j < 8; ++j)
                sH[srow + j][nm] = h[j];
        }
        __syncthreads();

        // Vertical pass: Out(16x16) = W(16x28) * H(28x16), 7 K=4 WMMA steps.
        v8f c = {};
#pragma unroll
        for (int kc = 0; kc < 7; ++kc) {
            const int k0 = 4 * kc + half;
            v2f b;
            b[0] = sH[k0][nm];
            b[1] = sH[k0 + 1][nm];
            c = __builtin_amdgcn_wmma_f32_16x16x4_f32(
                    false, wchunk[kc], false, b, (short)0, c, false, false);
        }
        acc[p] = c;
        __syncthreads();   // protect sH before next product overwrites it
    }

    // ---- SSIM map + mean reduction ----------------------------------------
    const float inv_total = 1.0f / (float)(NPLANES * IMG_H * IMG_W);
    float s = 0.0f;
#pragma unroll
    for (int j = 0; j < 8; ++j) {
        const float m1  = acc[0][j];
        const float m2  = acc[1][j];
        const float m1s = m1 * m1;
        const float m2s = m2 * m2;
        const float m12 = m1 * m2;
        const float s1  = acc[2][j] - m1s;
        const float s2  = acc[3][j] - m2s;
        const float s12 = acc[4][j] - m12;
        const float num = (2.0f * m12 + C1_SSIM) * (2.0f * s12 + C2_SSIM);
        const float den = (m1s + m2s + C1_SSIM) * (s1 + s2 + C2_SSIM);
        s += num / den;
    }
    s *= inv_total;

    // Wave32 reduction, then one atomic per tile.
#pragma unroll
    for (int off = 16; off > 0; off >>= 1)
        s += __shfl_down(s, off, 32);
    if (lane == 0)
        atomicAdd(out, s);
}

extern "C" void kernel_launch(void* const* d_in, const int* in_sizes, int n_in,
                              void* d_out, int out_size, void* d_ws, size_t ws_size,
                              hipStream_t stream) {
    (void)in_sizes; (void)n_in; (void)d_ws; (void)ws_size; (void)out_size;
    const float* img1 = (const float*)d_in[0];
    const float* img2 = (const float*)d_in[1];
    float* out = (float*)d_out;

    ssim_zero_kernel<<<dim3(1), dim3(32), 0, stream>>>(out);
    // 48 planes * 32*32 tiles = 49152 waves, one 16x16 tile each.
    ssim_wmma_kernel<<<dim3(NPLANES * 32 * 32), dim3(32), 0, stream>>>(img1, img2, out);
}